// GCN_13769665151543
// MI455X (gfx1250) — compile-verified
//
#include <hip/hip_runtime.h>
#include <hip/hip_bf16.h>
#include <math.h>

// ---------------------------------------------------------------------------
// GCN 2-layer forward for MI455X (gfx1250, wave32).
// Memory-bound workload (edge gather/scatter ~1.6 GB total >> 2.5 GFLOP of
// GEMM), so GEMMs use full-precision V_WMMA_F32_16X16X4_F32 to match the
// fp32 reference while still running on the CDNA5 matrix pipe.
// ---------------------------------------------------------------------------

typedef __attribute__((ext_vector_type(2))) float v2f;
typedef __attribute__((ext_vector_type(8))) float v8f;

// ---------------- degree / norm -------------------------------------------
__global__ __launch_bounds__(256) void gcn_deg_kernel(
    const int* __restrict__ src, const int* __restrict__ dst,
    float* __restrict__ degO, float* __restrict__ degI, int E)
{
    int t = blockIdx.x * 256 + threadIdx.x;
    if (t < E) {
        atomicAdd(&degO[src[t]], 1.0f);
        atomicAdd(&degI[dst[t]], 1.0f);
    }
}

__global__ __launch_bounds__(256) void gcn_norm_kernel(float* __restrict__ d, int n)
{
    int t = blockIdx.x * 256 + threadIdx.x;
    if (t < n) d[t] = rsqrtf(fmaxf(d[t], 1.0f));
}

// ---------------- WMMA GEMM: out[N,64] = (X * normS[:,None]) @ W[K,64] -----
// Block: 256 threads = 8 wave32 waves. Block tile: 32 rows x 64 cols.
// Wave w: rows (w>>2)*16..+15, cols (w&3)*16..+15; one v_wmma_f32_16x16x4_f32
// per K-step of 4. fp32 fragment layouts per CDNA5 ISA 7.12.2:
//   A 16x4 : lane g=lane>>4, M=lane&15 ; a[v] = A[M][k0 + 2g + v]
//   B 4x16 : b[v] = B[k0 + 2g + v][N=lane&15]
//   D 16x16: vgpr r -> row (g*8 + r), col lane&15
template <int K>
__global__ __launch_bounds__(256) void gcn_gemm_wmma(
    const float* __restrict__ X, const float* __restrict__ normS,
    const float* __restrict__ W, float* __restrict__ out, int nrows)
{
    __shared__ float sW[K * 64];   // K x 64, row-major
    __shared__ float sA[32 * K];   // 32 x K, row-major, pre-scaled by normS

    const int tid = threadIdx.x;
    const int rowTile = blockIdx.x * 32;

    for (int i = tid; i < K * 64; i += 256) sW[i] = W[i];
    for (int i = tid; i < 32 * K; i += 256) {
        int r = i / K, c = i - r * K;
        int row = rowTile + r;
        if (row >= nrows) row = nrows - 1;          // clamp (N%32==0 in practice)
        sA[i] = X[(size_t)row * K + c] * normS[row];
    }
    __syncthreads();

    const int wave = tid >> 5;
    const int lane = tid & 31;
    const int g    = lane >> 4;     // half-wave select
    const int ln   = lane & 15;
    const int rowHalf = wave >> 2;            // 0 or 1
    const int colBase = (wave & 3) << 4;      // 0,16,32,48
    const int mrow = rowHalf * 16 + ln;       // A row within the 32-row tile

    v8f acc = {0.f, 0.f, 0.f, 0.f, 0.f, 0.f, 0.f, 0.f};
#pragma unroll
    for (int k0 = 0; k0 < K; k0 += 4) {
        v2f a, b;
        a.x = sA[mrow * K + k0 + 2 * g + 0];
        a.y = sA[mrow * K + k0 + 2 * g + 1];
        b.x = sW[(k0 + 2 * g + 0) * 64 + colBase + ln];
        b.y = sW[(k0 + 2 * g + 1) * 64 + colBase + ln];
        acc = __builtin_amdgcn_wmma_f32_16x16x4_f32(
            /*neg_a=*/false, a, /*neg_b=*/false, b,
            /*c_mod=*/(short)0, acc, /*reuse_a=*/false, /*reuse_b=*/false);
    }

    const int rowBase = rowTile + rowHalf * 16;
#pragma unroll
    for (int r = 0; r < 8; ++r) {
        int row = rowBase + g * 8 + r;
        if (row < nrows)
            out[(size_t)row * 64 + colBase + ln] = acc[r];
    }
}

// ---------------- edge scatter-add: agg[dst] += msgs[src] ------------------
// 16 threads per edge; each thread moves one float4 (coalesced gather,
// 4 global float atomics for the segment-sum).
__global__ __launch_bounds__(256) void gcn_scatter_kernel(
    const float* __restrict__ msgs, const int* __restrict__ src,
    const int* __restrict__ dst, float* __restrict__ agg, int E)
{
    long long t = (long long)blockIdx.x * 256 + threadIdx.x;
    long long e = t >> 4;
    if (e >= E) return;
    int q = (int)(t & 15);
    int s = src[e], d = dst[e];
    const float4 v = *reinterpret_cast<const float4*>(msgs + (size_t)s * 64 + q * 4);
    float* p = agg + (size_t)d * 64 + q * 4;
    atomicAdd(p + 0, v.x);
    atomicAdd(p + 1, v.y);
    atomicAdd(p + 2, v.z);
    atomicAdd(p + 3, v.w);
}

// ---------------- agg*norm_dst + b -> ELU, accumulate BN stats -------------
// Block: 256 threads = 4 rows x 64 feats. LDS float atomics (ds_add_f32)
// reduce per-feature sum / sumsq, then one global atomic per feature.
__global__ __launch_bounds__(256) void gcn_post_kernel(
    const float* __restrict__ agg, const float* __restrict__ normD,
    const float* __restrict__ bias, float* __restrict__ out,
    float* __restrict__ stats, int n)
{
    __shared__ float ssum[64];
    __shared__ float ssq[64];
    int tid = threadIdx.x;
    if (tid < 64) { ssum[tid] = 0.f; ssq[tid] = 0.f; }
    __syncthreads();

    int f = tid & 63;
    int row = blockIdx.x * 4 + (tid >> 6);
    if (row < n) {
        float x = agg[(size_t)row * 64 + f] * normD[row] + bias[f];
        float y = x > 0.f ? x : (expf(x) - 1.0f);   // ELU(alpha=1)
        out[(size_t)row * 64 + f] = y;
        atomicAdd(&ssum[f], y);
        atomicAdd(&ssq[f], y * y);
    }
    __syncthreads();
    if (tid < 64) {
        atomicAdd(&stats[tid], ssum[tid]);
        atomicAdd(&stats[64 + tid], ssq[tid]);
    }
}

// ---------------- BatchNorm apply ------------------------------------------
__global__ __launch_bounds__(256) void gcn_bn_kernel(
    const float* __restrict__ in, const float* __restrict__ stats,
    const float* __restrict__ gamma, const float* __restrict__ beta,
    float* __restrict__ out, long long total, float invN)
{
    long long t = (long long)blockIdx.x * 256 + threadIdx.x;
    if (t >= total) return;
    int f = (int)(t & 63);
    float mean = stats[f] * invN;
    float var  = stats[64 + f] * invN - mean * mean;
    out[t] = (in[t] - mean) * rsqrtf(var + 1e-5f) * gamma[f] + beta[f];
}

// ---------------------------------------------------------------------------
extern "C" void kernel_launch(void* const* d_in, const int* in_sizes, int n_in,
                              void* d_out, int out_size, void* d_ws, size_t ws_size,
                              hipStream_t stream)
{
    const float* feat = (const float*)d_in[0];   // [N,128]
    const float* W1   = (const float*)d_in[1];   // [128,64]
    const float* b1   = (const float*)d_in[2];   // [64]
    const float* g1   = (const float*)d_in[3];
    const float* be1  = (const float*)d_in[4];
    const float* W2   = (const float*)d_in[5];   // [64,64]
    const float* b2   = (const float*)d_in[6];
    const float* g2   = (const float*)d_in[7];
    const float* be2  = (const float*)d_in[8];
    const int*   src  = (const int*)d_in[9];     // [E]
    const int*   dst  = (const int*)d_in[10];    // [E]

    const int N = in_sizes[0] / 128;
    const int E = in_sizes[9];
    const float invN = 1.0f / (float)N;

    // Workspace layout (floats):
    //   [0,N)            deg_out -> norm_src
    //   [N,2N)           deg_in  -> norm_dst
    //   [2N,2N+128)      BN stats (sum[64], sumsq[64])
    //   bufA [N*64], bufB [N*64]
    float* ws    = (float*)d_ws;
    float* normS = ws;
    float* normD = ws + (size_t)N;
    float* stats = ws + 2 * (size_t)N;
    float* bufA  = stats + 128;
    float* bufB  = bufA + (size_t)N * 64;

    const int blkE   = (E + 255) / 256;
    const int blkN2  = (2 * N + 255) / 256;
    const int blkGemm = (N + 31) / 32;
    const int blkScat = (int)(((long long)E * 16 + 255) / 256);
    const int blkPost = (N + 3) / 4;
    const long long total = (long long)N * 64;
    const int blkBN  = (int)((total + 255) / 256);

    // --- degrees & norms ---
    hipMemsetAsync(normS, 0, sizeof(float) * 2 * (size_t)N, stream);
    hipMemsetAsync(stats, 0, sizeof(float) * 128, stream);
    hipMemsetAsync(bufB, 0, sizeof(float) * (size_t)N * 64, stream);
    gcn_deg_kernel<<<blkE, 256, 0, stream>>>(src, dst, normS, normD, E);
    gcn_norm_kernel<<<blkN2, 256, 0, stream>>>(normS, 2 * N);

    // --- layer 1 ---
    gcn_gemm_wmma<128><<<blkGemm, 256, 0, stream>>>(feat, normS, W1, bufA, N);
    gcn_scatter_kernel<<<blkScat, 256, 0, stream>>>(bufA, src, dst, bufB, E);
    gcn_post_kernel<<<blkPost, 256, 0, stream>>>(bufB, normD, b1, bufA, stats, N);
    gcn_bn_kernel<<<blkBN, 256, 0, stream>>>(bufA, stats, g1, be1, bufB, total, invN);
    hipMemsetAsync(stats, 0, sizeof(float) * 128, stream);

    // --- layer 2 ---
    gcn_gemm_wmma<64><<<blkGemm, 256, 0, stream>>>(bufB, normS, W2, bufA, N);
    hipMemsetAsync(bufB, 0, sizeof(float) * (size_t)N * 64, stream);  // agg buffer
    gcn_scatter_kernel<<<blkScat, 256, 0, stream>>>(bufA, src, dst, bufB, E);
    gcn_post_kernel<<<blkPost, 256, 0, stream>>>(bufB, normD, b2, bufA, stats, N);
    gcn_bn_kernel<<<blkBN, 256, 0, stream>>>(bufA, stats, g2, be2, (float*)d_out, total, invN);
}